// DZSpecimenClfToy_22007412425146
// MI455X (gfx1250) — compile-verified
//
#include <hip/hip_runtime.h>
#include <hip/hip_bf16.h>
#include <math.h>

typedef __attribute__((ext_vector_type(2))) float v2f;
typedef __attribute__((ext_vector_type(8))) float v8f;

#define BATCH 8
#define NPTS  4096
#define PSZ   4
#define NLOG  (NPTS * 2)                 // 8192 logits per batch
#define KDIM  48
#define PATCH_FLAT (NPTS * PSZ * PSZ * 3) // 196608

// workspace layout (float offsets)
#define WS_FLAT 0                         // 8*48 resized topview
#define WS_XY   512                       // 8*8192 pixel coords (even=x/row, odd=y/col)
#define WS_PART (512 + BATCH * NLOG)      // 8*16*2 partial sums

// ---------------------------------------------------------------------------
// Kernel 1: 64x64 -> 4x4 bilinear resize (exact 2x2 block means at half-pixel
// sample points 7.5/23.5/39.5/55.5). One thread per (b,c,i,j) output.
// ---------------------------------------------------------------------------
__global__ void k_resize(const float* __restrict__ topview, float* __restrict__ ws) {
  int t = threadIdx.x;                  // 0..383
  int b = t / KDIM, idx = t % KDIM;
  int c = idx >> 4, rem = idx & 15, i = rem >> 2, j = rem & 3;
  int ri = 16 * i + 7, cj = 16 * j + 7;
  const float* img = topview + ((size_t)(b * 3 + c)) * 64 * 64;
  float s = img[ri * 64 + cj] + img[ri * 64 + cj + 1] +
            img[(ri + 1) * 64 + cj] + img[(ri + 1) * 64 + cj + 1];
  ws[WS_FLAT + t] = 0.25f * s;
}

// ---------------------------------------------------------------------------
// Kernel 2: coords = sigmoid(flat(8x48) @ W1^T(48x8192) + b1) * pad + 2
// Full-f32 WMMA: V_WMMA_F32_16X16X4_F32, one 16x16 output tile per wave,
// 12 k-steps of K=4.  M=16 pads the 8 batch rows with zeros (LDS-padded A).
// ---------------------------------------------------------------------------
__global__ void k_coords(const float* __restrict__ ws_in,
                         float* __restrict__ ws_xy,
                         const float* __restrict__ W1,
                         const float* __restrict__ b1,
                         const int* __restrict__ svh_p,
                         const int* __restrict__ svw_p) {
  __shared__ float sflat[16 * KDIM];    // rows 8..15 zero-padded
  for (int t = threadIdx.x; t < 16 * KDIM; t += blockDim.x) {
    int row = t / KDIM, col = t % KDIM;
    sflat[t] = (row < BATCH) ? ws_in[WS_FLAT + row * KDIM + col] : 0.0f;
  }
  __syncthreads();

  const float pad_h = (float)(svh_p[0] - 1 - PSZ);
  const float pad_w = (float)(svw_p[0] - 1 - PSZ);

  int wid  = blockIdx.x * (blockDim.x >> 5) + (threadIdx.x >> 5); // global wave
  int lane = threadIdx.x & 31;
  int half = lane >> 4;                 // 0: K+{0,1}, 1: K+{2,3}
  int lm   = lane & 15;
  int n0   = wid * 16;
  int n    = n0 + lm;                   // logit column handled by this lane
  const float* w1row = W1 + (size_t)n * KDIM;

  v8f acc = {};
#pragma unroll
  for (int kk = 0; kk < 12; ++kk) {
    int koff = kk * 4 + half * 2;
    v2f a;  a.x  = sflat[lm * KDIM + koff];  a.y  = sflat[lm * KDIM + koff + 1];
    v2f bm; bm.x = w1row[koff];              bm.y = w1row[koff + 1];
    acc = __builtin_amdgcn_wmma_f32_16x16x4_f32(
        /*neg_a=*/false, a, /*neg_b=*/false, bm,
        /*c_mod=*/(short)0, acc, /*reuse_a=*/false, /*reuse_b=*/false);
  }

  // D layout: VGPR r -> M=r for lanes 0-15; batch rows 0..7 all live there.
  if (half == 0) {
    float b1v   = b1[n];
    float scale = (n & 1) ? pad_w : pad_h;  // even index = x (rows), odd = y (cols)
#pragma unroll
    for (int r = 0; r < BATCH; ++r) {
      float logit = acc[r] + b1v;
      float sg    = 1.0f / (1.0f + expf(-logit));
      ws_xy[(size_t)r * NLOG + n] = sg * scale + (float)(PSZ / 2);
    }
  }
}

// ---------------------------------------------------------------------------
// Kernel 3: per-point 5x5x3 bilinear gather fused with the W2 dot-product.
// Each thread owns one (b, point): 75 image loads issued up-front, 48 lerped
// patch values, 96 coalesced W2 reads -> two scalar accumulators, then a
// deterministic LDS tree reduction to one partial pair per block.
// ---------------------------------------------------------------------------
__global__ void k_gather(const float* __restrict__ sv,
                         const float* __restrict__ W2,
                         const float* __restrict__ ws_xy,
                         float* __restrict__ part,
                         const int* __restrict__ svh_p,
                         const int* __restrict__ svw_p) {
  const int b  = blockIdx.y;
  const int p  = blockIdx.x * blockDim.x + threadIdx.x;   // point id 0..4095
  const int Hh = svh_p[0], Ww = svw_p[0];
  const float* img = sv + (size_t)b * Hh * Ww * 3;

  float x = ws_xy[(size_t)b * NLOG + 2 * p];
  float y = ws_xy[(size_t)b * NLOG + 2 * p + 1];
  float tlr = x - (float)(PSZ / 2), tlc = y - (float)(PSZ / 2);
  float r0f = floorf(tlr), c0f = floorf(tlc);
  float fr = tlr - r0f, fc = tlc - c0f;
  int r0 = (int)r0f, c0 = (int)c0f;

  int rr[5], cc[5];
#pragma unroll
  for (int k = 0; k < 5; ++k) {
    rr[k] = min(max(r0 + k, 0), Hh - 1);
    cc[k] = min(max(c0 + k, 0), Ww - 1);
  }

  // Issue all 75 loads before any math (single loadcnt drain).
  float V[5][5][3];
#pragma unroll
  for (int i = 0; i < 5; ++i) {
    const float* rowp = img + (size_t)rr[i] * Ww * 3;
#pragma unroll
    for (int j = 0; j < 5; ++j) {
      const float* pix = rowp + cc[j] * 3;
      V[i][j][0] = pix[0]; V[i][j][1] = pix[1]; V[i][j][2] = pix[2];
    }
  }

  const float* w2a = W2 + (size_t)p * KDIM;
  const float* w2b = W2 + (size_t)PATCH_FLAT + (size_t)p * KDIM;
  float gr = 1.0f - fr, gc = 1.0f - fc;
  float acc0 = 0.0f, acc1 = 0.0f;
#pragma unroll
  for (int i = 0; i < PSZ; ++i) {
#pragma unroll
    for (int j = 0; j < PSZ; ++j) {
#pragma unroll
      for (int ch = 0; ch < 3; ++ch) {
        float top = V[i][j][ch] * gc + V[i][j + 1][ch] * fc;
        float bot = V[i + 1][j][ch] * gc + V[i + 1][j + 1][ch] * fc;
        float val = top * gr + bot * fr;
        int f = i * 12 + j * 3 + ch;
        acc0 = fmaf(w2a[f], val, acc0);
        acc1 = fmaf(w2b[f], val, acc1);
      }
    }
  }

  __shared__ float red0[256];
  __shared__ float red1[256];
  red0[threadIdx.x] = acc0;
  red1[threadIdx.x] = acc1;
  __syncthreads();
#pragma unroll
  for (int s = 128; s > 0; s >>= 1) {
    if ((int)threadIdx.x < s) {
      red0[threadIdx.x] += red0[threadIdx.x + s];
      red1[threadIdx.x] += red1[threadIdx.x + s];
    }
    __syncthreads();
  }
  if (threadIdx.x == 0) {
    part[(b * gridDim.x + blockIdx.x) * 2 + 0] = red0[0];
    part[(b * gridDim.x + blockIdx.x) * 2 + 1] = red1[0];
  }
}

// ---------------------------------------------------------------------------
// Kernel 4: fixed-order final reduction (deterministic, no float atomics).
// ---------------------------------------------------------------------------
__global__ void k_final(const float* __restrict__ part,
                        const float* __restrict__ b2,
                        float* __restrict__ out, int nblk) {
  int t = threadIdx.x;                  // 0..15 -> (b,k)
  if (t < BATCH * 2) {
    int b = t >> 1, k = t & 1;
    float s = b2[k];
    for (int blk = 0; blk < nblk; ++blk)
      s += part[(b * nblk + blk) * 2 + k];
    out[b * 2 + k] = s;
  }
}

extern "C" void kernel_launch(void* const* d_in, const int* in_sizes, int n_in,
                              void* d_out, int out_size, void* d_ws, size_t ws_size,
                              hipStream_t stream) {
  const float* topview = (const float*)d_in[0];
  const float* sv      = (const float*)d_in[1];
  const float* W1      = (const float*)d_in[2];
  const float* b1      = (const float*)d_in[3];
  const float* W2      = (const float*)d_in[4];
  const float* b2      = (const float*)d_in[5];
  const int*   svh     = (const int*)d_in[6];
  const int*   svw     = (const int*)d_in[7];
  float* ws  = (float*)d_ws;
  float* out = (float*)d_out;

  // 1) topview resize -> ws[WS_FLAT]
  k_resize<<<1, BATCH * KDIM, 0, stream>>>(topview, ws);

  // 2) WMMA f32 GEMM + sigmoid + scale -> ws[WS_XY]
  //    512 tiles of 16 logit-columns: 64 blocks x 8 waves.
  k_coords<<<64, 256, 0, stream>>>(ws, ws + WS_XY, W1, b1, svh, svw);

  // 3) fused gather + W2 dot: 16 blocks of 256 points per batch.
  k_gather<<<dim3(NPTS / 256, BATCH), 256, 0, stream>>>(
      sv, W2, ws + WS_XY, ws + WS_PART, svh, svw);

  // 4) deterministic final reduce (+ bias) -> d_out (8x2)
  k_final<<<1, 32, 0, stream>>>(ws + WS_PART, b2, out, NPTS / 256);
}